// KnnMutualInformation_86844238725479
// MI455X (gfx1250) — compile-verified
//
#include <hip/hip_runtime.h>
#include <math.h>

// KSG k-NN mutual information, B=2,C=2,H=W=64 -> BC=4 channels, HW=4096 points.
// Per channel: eps_i = 6th smallest max(|xi-xj|,|yi-yj|); nx_i = #{j: |xi-xj|<eps_i};
// mi_c = dg(5)+dg(4096) - mean_i(dg(nx_i)+dg(ny_i)), clamped at 0.

#define HW   4096
#define KNN  5
#define TPB  256

typedef unsigned int u32x4 __attribute__((ext_vector_type(4)));
typedef int          i32x4 __attribute__((ext_vector_type(4)));
typedef int          i32x8 __attribute__((ext_vector_type(8)));

#if defined(__HIP_DEVICE_COMPILE__) && __has_builtin(__builtin_amdgcn_tensor_load_to_lds)
#define USE_TDM 1
#else
#define USE_TDM 0
#endif

// digamma for positive args (here: integers in [1,4096] and the constants 5, 4096).
// Shift-up recurrence to x>=6, then asymptotic series: psi(x) ~ ln x - 1/2x - 1/12x^2 + ...
__device__ __forceinline__ float dev_digamma(float x) {
  float r = 0.0f;
  while (x < 6.0f) { r -= 1.0f / x; x += 1.0f; }
  float inv  = 1.0f / x;
  float inv2 = inv * inv;
  float ser  = inv2 * (1.0f / 12.0f - inv2 * (1.0f / 120.0f - inv2 * (1.0f / 252.0f)));
  return r + logf(x) - 0.5f * inv - ser;
}

__device__ __forceinline__ unsigned lds_addr_of(const void* p) {
  // Generic pointer to LDS: addr[63:32] = shared aperture, addr[31:0] = LDS byte offset.
  return (unsigned)(unsigned long long)p;
}

#if USE_TDM
// Build a D# (ISA 8.3-8.6) for a contiguous row of n fp32 elements and issue TDM.
// Group0: count=1 | lds_addr | global_addr[31:0] | global_addr[56:32] + type=2.
// Group1: data_size=4B; tensor_dim0=n; tensor_dim1=1; tile_dim0=n; tile_dim1=1;
//         tensor_dim0_stride=n; no padding, no multicast, no iterate.
__device__ __forceinline__ void tdm_load_row(unsigned lds_byte_addr, const float* gptr, int n) {
  unsigned long long ga = (unsigned long long)gptr;
  u32x4 g0;
  g0[0] = 1u;                                                 // count=1 (valid user D#)
  g0[1] = lds_byte_addr;                                      // lds_addr
  g0[2] = (unsigned)(ga & 0xFFFFFFFFull);                     // global_addr[31:0]
  g0[3] = (unsigned)((ga >> 32) & 0x01FFFFFFull) | (2u << 30);// global_addr[56:32] | type=2
  i32x8 g1;
  g1[0] = (int)(2u << 16);                                    // data_size=2 -> 4 bytes
  g1[1] = (int)((unsigned)n << 16);                           // tensor_dim0[15:0] @ bits63:48
  g1[2] = (int)(1u << 16);                                    // tensor_dim1=1    @ bits95:80
  g1[3] = (int)((unsigned)n << 16);                           // tile_dim0=n      @ bits127:112
  g1[4] = 1;                                                  // tile_dim1=1 (tile_dim2=0)
  g1[5] = n;                                                  // tensor_dim0_stride[31:0]
  g1[6] = 0;                                                  // stride hi / dim1_stride lo
  g1[7] = 0;                                                  // dim1_stride hi
  i32x4 gz = {0, 0, 0, 0};                                    // groups 2/3 unused (2D tensor)
#if defined(__clang_major__) && (__clang_major__ >= 23)
  i32x8 gz8 = {0, 0, 0, 0, 0, 0, 0, 0};
  __builtin_amdgcn_tensor_load_to_lds(g0, g1, gz, gz, gz8, 0);
#else
  __builtin_amdgcn_tensor_load_to_lds(g0, g1, gz, gz, 0);
#endif
}
#endif

__global__ void __launch_bounds__(TPB)
knnmi_kernel(const float* __restrict__ x, const float* __restrict__ y,
             float* __restrict__ acc) {
  __shared__ __align__(16) float xs[HW];
  __shared__ __align__(16) float ys[HW];
  __shared__ float red[TPB];

  const int tid = threadIdx.x;
  const int c   = blockIdx.y;
  const float* xc = x + (size_t)c * HW;
  const float* yc = y + (size_t)c * HW;

#if USE_TDM
  if (tid < 32) {                       // one wave issues both DMAs (EXEC ignored by TDM)
    tdm_load_row(lds_addr_of(xs), xc, HW);
    tdm_load_row(lds_addr_of(ys), yc, HW);
#if __has_builtin(__builtin_amdgcn_s_wait_tensorcnt)
    __builtin_amdgcn_s_wait_tensorcnt(0);
#else
    asm volatile("s_wait_tensorcnt 0" ::: "memory");
#endif
  }
#else
  for (int j = tid; j < HW; j += TPB) { xs[j] = xc[j]; ys[j] = yc[j]; }
#endif
  __syncthreads();

  const int   i  = blockIdx.x * TPB + tid;
  const float xi = xs[i];
  const float yi = ys[i];

  // Pass 1: running 6 smallest joint distances (ascending s0..s5), branch-guarded
  // min/max insertion network. Self distance 0 is included, matching the reference.
  float s0 = 3.0e38f, s1 = 3.0e38f, s2 = 3.0e38f,
        s3 = 3.0e38f, s4 = 3.0e38f, s5 = 3.0e38f;

  const float4* xs4 = (const float4*)xs;
  const float4* ys4 = (const float4*)ys;

#define KNN_PROC(ax, bx)                                           \
  {                                                                \
    float z = fmaxf(fabsf(xi - (ax)), fabsf(yi - (bx)));           \
    if (z < s5) {                                                  \
      float t = z, m;                                              \
      m = fminf(s0, t); t = fmaxf(s0, t); s0 = m;                  \
      m = fminf(s1, t); t = fmaxf(s1, t); s1 = m;                  \
      m = fminf(s2, t); t = fmaxf(s2, t); s2 = m;                  \
      m = fminf(s3, t); t = fmaxf(s3, t); s3 = m;                  \
      m = fminf(s4, t); t = fmaxf(s4, t); s4 = m;                  \
      s5 = fminf(s5, t);                                           \
    }                                                              \
  }

  for (int j4 = 0; j4 < HW / 4; ++j4) {
    float4 a = xs4[j4];   // lane-uniform ds_load_b128 broadcast
    float4 b = ys4[j4];
    KNN_PROC(a.x, b.x);
    KNN_PROC(a.y, b.y);
    KNN_PROC(a.z, b.z);
    KNN_PROC(a.w, b.w);
  }
#undef KNN_PROC

  const float eps = s5;   // (k+1)-th smallest joint distance

  // Pass 2: strict-threshold marginal neighbor counts (self included when eps>0).
  float nx = 0.0f, ny = 0.0f;
  for (int j4 = 0; j4 < HW / 4; ++j4) {
    float4 a = xs4[j4];
    float4 b = ys4[j4];
    nx += (fabsf(xi - a.x) < eps) ? 1.0f : 0.0f;
    ny += (fabsf(yi - b.x) < eps) ? 1.0f : 0.0f;
    nx += (fabsf(xi - a.y) < eps) ? 1.0f : 0.0f;
    ny += (fabsf(yi - b.y) < eps) ? 1.0f : 0.0f;
    nx += (fabsf(xi - a.z) < eps) ? 1.0f : 0.0f;
    ny += (fabsf(yi - b.z) < eps) ? 1.0f : 0.0f;
    nx += (fabsf(xi - a.w) < eps) ? 1.0f : 0.0f;
    ny += (fabsf(yi - b.w) < eps) ? 1.0f : 0.0f;
  }

  float v = dev_digamma(nx) + dev_digamma(ny);

  // Block tree reduction, then one float atomic per block into the channel accumulator.
  red[tid] = v;
  __syncthreads();
  for (int s = TPB / 2; s > 0; s >>= 1) {
    if (tid < s) red[tid] += red[tid + s];
    __syncthreads();
  }
  if (tid == 0) atomicAdd(&acc[c], red[0]);
}

__global__ void knnmi_finalize(const float* __restrict__ acc, float* __restrict__ out, int bc) {
  int i = threadIdx.x;
  if (i < bc) {
    float cst = dev_digamma((float)KNN) + dev_digamma((float)HW);
    float mi  = cst - acc[i] * (1.0f / (float)HW);
    out[i] = fmaxf(mi, 0.0f);
  }
}

extern "C" void kernel_launch(void* const* d_in, const int* in_sizes, int n_in,
                              void* d_out, int out_size, void* d_ws, size_t ws_size,
                              hipStream_t stream) {
  const float* x = (const float*)d_in[0];
  const float* y = (const float*)d_in[1];
  float* out = (float*)d_out;
  float* acc = (float*)d_ws;

  const int bc = in_sizes[0] / HW;  // = 4 channels

  // d_ws is poisoned, not re-zeroed between replays: zero the accumulators every call
  // (hipMemsetAsync is graph-capturable).
  hipMemsetAsync(acc, 0, (size_t)bc * sizeof(float), stream);

  dim3 grid(HW / TPB, bc);
  knnmi_kernel<<<grid, TPB, 0, stream>>>(x, y, acc);
  knnmi_finalize<<<1, 32, 0, stream>>>(acc, out, bc);
}